// ResidualVectorQuantizer_59605556134140
// MI455X (gfx1250) — compile-verified
//
#include <hip/hip_runtime.h>
#include <hip/hip_bf16.h>

// ResidualVectorQuantizer fused kernel for gfx1250 (MI455X).
// N=65536, D=512, K=256, S=4, BETA=0.25.
// FP32 WMMA (v_wmma_f32_16x16x4_f32) for the distance GEMMs.
// Residual tile resident in LDS across all 4 stages; codebook chunks staged
// through a shared LDS buffer so each block reads the codebook once per stage.
// Dual accumulator chains (even/odd d-step) double WMMA ILP and avoid
// register-rotation copies across the pipelined chunk loop.

typedef __attribute__((ext_vector_type(2))) float v2f;
typedef __attribute__((ext_vector_type(4))) float v4f;
typedef __attribute__((ext_vector_type(8))) float v8f;

#define RVQ_D 512
#define RVQ_K 256
#define RVQ_S 4
#define RVQ_TM 64        // rows per block
#define RVQ_DB 8         // d-columns per staged codebook chunk
#define RVQ_NC (RVQ_D / RVQ_DB)   // 64 chunks per stage
#define RVQ_BSTRIDE 10   // padded LDS row stride for Bbuf (floats): 8B-aligned,
                         // bank stride 10 -> conflict-free for lane deltas < 32

// ---------------------------------------------------------------------------
// Kernel 1: codebook row norms  ||w_k||^2  -> ws[0 .. S*K)
// ---------------------------------------------------------------------------
__global__ __launch_bounds__(256) void rvq_wk2_kernel(const float* __restrict__ cb,
                                                      float* __restrict__ wk2,
                                                      int total) {
    int t = blockIdx.x * blockDim.x + threadIdx.x;
    if (t < total) {
        const float* w = cb + (size_t)t * RVQ_D;
        float s = 0.f;
        #pragma unroll 8
        for (int j = 0; j < RVQ_D; ++j) s += w[j] * w[j];
        wk2[t] = s;
    }
}

// ---------------------------------------------------------------------------
// Kernel 2: main fused RVQ. One block = 64 rows, all 4 stages.
// Dynamic LDS = 160,256 B -> 2 workgroups per 320KB WGP.
// ---------------------------------------------------------------------------
__global__ __launch_bounds__(256) void rvq_main_kernel(
    const float* __restrict__ x,        // [N, D]
    const float* __restrict__ cb,       // [S, K, D]
    const float* __restrict__ wk2g,     // [S*K]
    float* __restrict__ xq_out,         // [N, D]
    int*   __restrict__ idx_out,        // [N, S]
    float* __restrict__ dist_out,       // [N, S, K]
    float* __restrict__ code_out,       // [N, S, D]
    float* __restrict__ blk_loss)       // [numBlocks, S] partial sums of r_new^2
{
    extern __shared__ char smem[];
    float* Rt    = (float*)smem;                    // [64][512] residual tile
    float* rn2   = Rt + RVQ_TM * RVQ_D;             // [64]   ||r_m||^2
    float* wk2s  = rn2 + RVQ_TM;                    // [256]  stage codebook norms
    float* candv = wk2s + RVQ_K;                    // [64*32] argmin candidate vals
    int*   candi = (int*)(candv + RVQ_TM * 32);     // [64*32] argmin candidate idxs
    int*   idxs  = candi + RVQ_TM * 32;             // [64]
    float* lsum  = (float*)(idxs + RVQ_TM);         // [256]
    float* Bbuf  = lsum + 256;                      // [256][RVQ_BSTRIDE] codebook chunk

    const int tid   = threadIdx.x;
    const int lane  = tid & 31;
    const int wv    = tid >> 5;         // 0..7 (8 wave32s)
    const int rt    = wv >> 1;          // row-tile 0..3 (16 rows each)
    const int khalf = wv & 1;           // which 128-wide k strip
    const int mr    = rt * 16;
    const int ln    = lane & 15;
    const int hi    = lane >> 4;        // 0: lanes 0-15, 1: lanes 16-31
    const int dsel  = hi * 2;           // fp32 16x16x4 A/B layout: upper half holds K=2,3
    const size_t m0 = (size_t)blockIdx.x * RVQ_TM;

    // ---- load residual tile (= x) into LDS, coalesced ----
    for (int i = tid; i < RVQ_TM * RVQ_D; i += 256)
        Rt[i] = x[m0 * RVQ_D + i];
    __syncthreads();

    // B-fragment LDS read pointers (fixed across stages/chunks)
    const float* bq[8];
    #pragma unroll
    for (int t = 0; t < 8; ++t)
        bq[t] = Bbuf + ((khalf * 8 + t) * 16 + ln) * RVQ_BSTRIDE + dsel;
    // A-fragment LDS read pointer (row base; chunk offset added in loop)
    const float* ap = Rt + (mr + ln) * RVQ_D + dsel;

    for (int s = 0; s < RVQ_S; ++s) {
        const float* Ws = cb + (size_t)s * RVQ_K * RVQ_D;
        const float* wrow = Ws + (size_t)tid * RVQ_D;   // thread t stages row t

        // ---- stage codebook norms -> LDS ----
        if (tid < RVQ_K) wk2s[tid] = wk2g[s * RVQ_K + tid];
        // ---- per-row residual norms ----
        if (tid < RVQ_TM) {
            const float* r = Rt + tid * RVQ_D;
            float acc = 0.f;
            #pragma unroll 8
            for (int j = 0; j < RVQ_D; ++j) acc += r[j] * r[j];
            rn2[tid] = acc;
        }

        // ---- GEMM: G[64,256] = R[64,512] @ W[256,512]^T via WMMA f32 16x16x4
        // Codebook streamed through LDS in 8-column chunks (read once / block).
        // Two independent accumulator chains (even/odd d-step) for ILP and
        // clean in-place D==C accumulation.
        v8f accE[8], accO[8];
        #pragma unroll
        for (int t = 0; t < 8; ++t) {
            accE[t] = (v8f){0.f, 0.f, 0.f, 0.f, 0.f, 0.f, 0.f, 0.f};
            accO[t] = (v8f){0.f, 0.f, 0.f, 0.f, 0.f, 0.f, 0.f, 0.f};
        }

        // register prefetch of chunk 0 (32B contiguous per thread)
        v4f pa = *(const v4f*)(wrow + 0);
        v4f pb = *(const v4f*)(wrow + 4);

        for (int c = 0; c < RVQ_NC; ++c) {
            __syncthreads();   // previous chunk fully consumed (also covers wk2s/rn2 on c==0)
            {   // stage prefetched chunk into Bbuf (8B-aligned, conflict-free stride)
                float* brow = Bbuf + tid * RVQ_BSTRIDE;
                *(v2f*)(brow + 0) = (v2f){pa.x, pa.y};
                *(v2f*)(brow + 2) = (v2f){pa.z, pa.w};
                *(v2f*)(brow + 4) = (v2f){pb.x, pb.y};
                *(v2f*)(brow + 6) = (v2f){pb.z, pb.w};
            }
            __syncthreads();   // Bbuf ready
            {   // unconditional prefetch of next chunk (wrapped); hides behind WMMAs
                const int cn = (c + 1) & (RVQ_NC - 1);
                pa = *(const v4f*)(wrow + cn * RVQ_DB);
                pb = *(const v4f*)(wrow + cn * RVQ_DB + 4);
            }
            const float* apc = ap + c * RVQ_DB;
            v2f a0 = *(const v2f*)(apc + 0);
            v2f a1 = *(const v2f*)(apc + 4);
            #pragma unroll
            for (int t = 0; t < 8; ++t) {
                v2f b0 = *(const v2f*)(bq[t] + 0);
                // (neg_a, A, neg_b, B, c_mod, C, reuse_a, reuse_b)
                accE[t] = __builtin_amdgcn_wmma_f32_16x16x4_f32(
                    false, a0, false, b0, (short)0, accE[t], false, false);
                v2f b1 = *(const v2f*)(bq[t] + 4);
                accO[t] = __builtin_amdgcn_wmma_f32_16x16x4_f32(
                    false, a1, false, b1, (short)0, accO[t], false, false);
            }
        }

        // ---- distances + per-lane argmin (C layout: VGPR j -> row j / j+8) ----
        float rr[8];
        #pragma unroll
        for (int j = 0; j < 8; ++j) rr[j] = rn2[mr + hi * 8 + j];

        float bestv[8]; int besti[8];
        #pragma unroll
        for (int j = 0; j < 8; ++j) { bestv[j] = 3.402823466e38f; besti[j] = 0; }

        #pragma unroll
        for (int t = 0; t < 8; ++t) {
            const int k = (khalf * 8 + t) * 16 + ln;
            const float w2 = wk2s[k];
            #pragma unroll
            for (int j = 0; j < 8; ++j) {
                const int m = mr + hi * 8 + j;
                const float g = accE[t][j] + accO[t][j];
                const float dist = rr[j] + w2 - 2.0f * g;
                dist_out[((m0 + m) * RVQ_S + s) * RVQ_K + k] = dist;
                // t ascending => k ascending within lane: strict '<' keeps first min
                if (dist < bestv[j]) { bestv[j] = dist; besti[j] = k; }
            }
        }
        #pragma unroll
        for (int j = 0; j < 8; ++j) {
            const int m = mr + hi * 8 + j;
            const int c = khalf * 16 + ln;      // 32 candidates per row
            candv[m * 32 + c] = bestv[j];
            candi[m * 32 + c] = besti[j];
        }
        __syncthreads();

        // ---- final argmin per row (first occurrence of min, like jnp.argmin) ----
        if (tid < RVQ_TM) {
            float bv = candv[tid * 32];
            int   bi = candi[tid * 32];
            for (int c = 1; c < 32; ++c) {
                float v = candv[tid * 32 + c];
                int   i = candi[tid * 32 + c];
                if (v < bv || (v == bv && i < bi)) { bv = v; bi = i; }
            }
            idxs[tid] = bi;
            idx_out[(m0 + tid) * RVQ_S + s] = bi;
        }
        __syncthreads();

        // ---- gather code, emit all_codes, update residual, loss partial ----
        // loss_s uses (code - r)^2 == (r_new)^2 since r_new = r - code.
        float part = 0.f;
        for (int i = tid; i < RVQ_TM * RVQ_D; i += 256) {
            const int m = i >> 9;
            const int dcol = i & (RVQ_D - 1);
            const float w = Ws[(size_t)idxs[m] * RVQ_D + dcol];
            code_out[((m0 + m) * RVQ_S + s) * RVQ_D + dcol] = w;
            const float r = Rt[i] - w;
            Rt[i] = r;
            part += r * r;
        }
        lsum[tid] = part;
        __syncthreads();
        for (int off = 128; off > 0; off >>= 1) {
            if (tid < off) lsum[tid] += lsum[tid + off];
            __syncthreads();
        }
        if (tid == 0) blk_loss[(size_t)blockIdx.x * RVQ_S + s] = lsum[0];
        __syncthreads();
    }

    // ---- x_q = x - final residual ----
    for (int i = tid; i < RVQ_TM * RVQ_D; i += 256) {
        const size_t g = m0 * RVQ_D + i;
        xq_out[g] = x[g] - Rt[i];
    }
}

// ---------------------------------------------------------------------------
// Kernel 3: deterministic loss reduction (fixed order every call).
// mean_loss = (1+BETA)/(S*N*D) * sum_{blocks,stages} partial
// ---------------------------------------------------------------------------
__global__ __launch_bounds__(256) void rvq_loss_kernel(const float* __restrict__ blk,
                                                       float* __restrict__ out,
                                                       int n, float scale) {
    __shared__ float sh[256];
    float p = 0.f;
    for (int i = threadIdx.x; i < n; i += 256) p += blk[i];
    sh[threadIdx.x] = p;
    __syncthreads();
    for (int off = 128; off > 0; off >>= 1) {
        if (threadIdx.x < off) sh[threadIdx.x] += sh[threadIdx.x + off];
        __syncthreads();
    }
    if (threadIdx.x == 0) out[0] = sh[0] * scale;
}

// ---------------------------------------------------------------------------
extern "C" void kernel_launch(void* const* d_in, const int* in_sizes, int n_in,
                              void* d_out, int out_size, void* d_ws, size_t ws_size,
                              hipStream_t stream) {
    const float* x  = (const float*)d_in[0];          // [N, D]
    const float* cb = (const float*)d_in[1];          // [S, K, D]
    const int N = in_sizes[0] / RVQ_D;                // 65536

    // d_out layout (return order): x_q [N,D] | mean_loss [1] | all_indices [N,S] (i32)
    //                              | all_distances [N,S,K] | all_codes [N,S,D]
    float* out   = (float*)d_out;
    float* xq    = out;
    float* loss  = out + (size_t)N * RVQ_D;
    int*   idxs  = (int*)(loss + 1);
    float* dists = (float*)(idxs + (size_t)N * RVQ_S);
    float* codes = dists + (size_t)N * RVQ_S * RVQ_K;

    // workspace: wk2 [S*K] | blk_loss [numBlocks*S]
    float* wk2      = (float*)d_ws;
    float* blk_loss = wk2 + RVQ_S * RVQ_K;

    const int NB = N / RVQ_TM;   // 1024 blocks

    rvq_wk2_kernel<<<(RVQ_S * RVQ_K + 255) / 256, 256, 0, stream>>>(cb, wk2, RVQ_S * RVQ_K);

    // LDS: Rt + rn2 + wk2s + candv + candi + idxs + lsum + Bbuf  = 160,256 B
    const size_t smem_bytes =
        (RVQ_TM * RVQ_D + RVQ_TM + RVQ_K + RVQ_TM * 32) * sizeof(float) +
        (RVQ_TM * 32 + RVQ_TM) * sizeof(int) +
        256 * sizeof(float) +
        RVQ_K * RVQ_BSTRIDE * sizeof(float);

    rvq_main_kernel<<<NB, 256, smem_bytes, stream>>>(x, cb, wk2, xq, idxs, dists,
                                                     codes, blk_loss);

    const float scale = (float)((1.0 + 0.25) /
                        ((double)RVQ_S * (double)N * (double)RVQ_D));
    rvq_loss_kernel<<<1, 256, 0, stream>>>(blk_loss, loss, NB * RVQ_S, scale);
}